// GCN_14559939134162
// MI455X (gfx1250) — compile-verified
//
#include <hip/hip_runtime.h>

typedef __attribute__((ext_vector_type(2))) float v2f;
typedef __attribute__((ext_vector_type(8))) float v8f;

#define TPB 256

// ---------------------------------------------------------------------------
// ws layout (floats), ns = n rounded up to 32 (so ntiles*16 <= ns and every
// array base is 128B-aligned):
//   [0,ns) deg | [ns,2ns) dinv | [2ns,3ns) s | [3ns,4ns) h2c0 | [4ns,5ns) h2c1
//   [5ns,6ns) agg0 | [6ns,7ns) agg1 | [7ns, 7ns+16n) relu1
// ---------------------------------------------------------------------------

__global__ void gcn_init(float* __restrict__ deg, float* __restrict__ s,
                         float* __restrict__ agg0, float* __restrict__ agg1,
                         int n) {
  int i = blockIdx.x * blockDim.x + threadIdx.x;
  if (i < n) {
    deg[i] = 1.0f;      // self-loop contributes 1 to every degree
    s[i] = 0.0f;
    agg0[i] = 0.0f;
    agg1[i] = 0.0f;
  }
}

__global__ void gcn_degree(const int* __restrict__ dst,
                           float* __restrict__ deg, int E) {
  int e = blockIdx.x * blockDim.x + threadIdx.x;
  if (e < E) unsafeAtomicAdd(&deg[dst[e]], 1.0f);
}

__global__ void gcn_dinv(const float* __restrict__ deg,
                         float* __restrict__ dinv, int n) {
  int i = blockIdx.x * blockDim.x + threadIdx.x;
  if (i < n) dinv[i] = rsqrtf(deg[i]);          // deg >= 1 always
}

// Layer 1 message is rank-1: h1[src] = x[src]*W1, so aggregate the scalar
// x[src]*norm per edge (1 atomic instead of 16) and expand by W1 afterwards.
__global__ void gcn_l1_scatter(const int* __restrict__ src,
                               const int* __restrict__ dst,
                               const float* __restrict__ x,
                               const float* __restrict__ dinv,
                               float* __restrict__ s, int E) {
  int e = blockIdx.x * blockDim.x + threadIdx.x;
  if (e >= E) return;
  int a = src[e], b = dst[e];
  unsafeAtomicAdd(&s[b], x[a] * dinv[a] * dinv[b]);
}

__global__ void gcn_l1_finalize(const float* __restrict__ x,
                                const float* __restrict__ dinv,
                                const float* __restrict__ s,
                                const float* __restrict__ W1,
                                const float* __restrict__ b1,
                                float* __restrict__ relu1, int n) {
  int i = blockIdx.x * blockDim.x + threadIdx.x;
  if (i >= n) return;
  float di = dinv[i];
  float sn = s[i] + x[i] * di * di;             // + self-loop (norm = 1/deg)
  float4* o = (float4*)(relu1 + (size_t)i * 16);
#pragma unroll
  for (int q = 0; q < 4; ++q) {
    float4 r;
    r.x = fmaxf(fmaf(sn, W1[4 * q + 0], b1[4 * q + 0]), 0.0f);
    r.y = fmaxf(fmaf(sn, W1[4 * q + 1], b1[4 * q + 1]), 0.0f);
    r.z = fmaxf(fmaf(sn, W1[4 * q + 2], b1[4 * q + 2]), 0.0f);
    r.w = fmaxf(fmaf(sn, W1[4 * q + 3], b1[4 * q + 3]), 0.0f);
    o[q] = r;
  }
}

// Layer-2 dense transform [16 nodes x 16] @ [16 x 2 (padded to 16)] done with
// four chained fp32 WMMAs (V_WMMA_F32_16X16X4_F32), one 16-node tile per wave.
// A layout (16x4 f32): lanes 0-15 hold K={4k,4k+1}, lanes 16-31 K={4k+2,4k+3}.
// B layout (4x16):     same K split, column = lane&15.
// D layout (16x16 f32): element [M = vgpr + 8*(lane>=16)][Ncol = lane&15].
//
// Branchless tail handling: A-rows past n are clamped to n-1 (each D row
// depends only on its own A row, so valid rows are exact), and the SoA output
// columns are padded to ns >= ntiles*16 so unconditional b128 stores of tail
// garbage land in never-read workspace slack.
__global__ void gcn_l2_dense_wmma(const float* __restrict__ relu1,
                                  const float* __restrict__ W2,
                                  float* __restrict__ h2c0,
                                  float* __restrict__ h2c1,
                                  int n, int ntiles) {
  int gtid = blockIdx.x * blockDim.x + threadIdx.x;
  int tile = gtid >> 5;
  int lane = threadIdx.x & 31;
  if (tile >= ntiles) return;                   // wave-uniform exit
  int base = tile * 16;
  int half = lane >> 4;
  int m = lane & 15;

  // B = W2 (row-major [16,2]) zero-padded to 16x16; column index = m.
  v2f bmat[4];
#pragma unroll
  for (int k = 0; k < 4; ++k) {
    int r0 = 4 * k + 2 * half;
    float bx = 0.0f, by = 0.0f;
    if (m < 2) { bx = W2[r0 * 2 + m]; by = W2[(r0 + 1) * 2 + m]; }
    bmat[k].x = bx; bmat[k].y = by;
  }

  // A = relu1 tile, row = base + m (clamped), cols {4k+2h, 4k+2h+1} per pair.
  int row = base + m;
  row = (row < n) ? row : (n - 1);
  const v2f* p = (const v2f*)(relu1 + (size_t)row * 16 + 2 * half);
  v2f amat[4];
#pragma unroll
  for (int k = 0; k < 4; ++k) amat[k] = p[2 * k];

  v8f c = {};
#pragma unroll
  for (int k = 0; k < 4; ++k)
    c = __builtin_amdgcn_wmma_f32_16x16x4_f32(
        /*neg_a=*/false, amat[k], /*neg_b=*/false, bmat[k],
        /*c_mod=*/(short)0, c, /*reuse_a=*/false, /*reuse_b=*/false);

  if (m < 2) {                                  // only class columns 0,1 real
    float* col = (m == 0) ? h2c0 : h2c1;
    int node0 = base + half * 8;                // 8 consecutive nodes per lane
    float4 lo; lo.x = c[0]; lo.y = c[1]; lo.z = c[2]; lo.w = c[3];
    float4 hi; hi.x = c[4]; hi.y = c[5]; hi.z = c[6]; hi.w = c[7];
    *(float4*)(col + node0)     = lo;
    *(float4*)(col + node0 + 4) = hi;
  }
}

__global__ void gcn_l2_scatter(const int* __restrict__ src,
                               const int* __restrict__ dst,
                               const float* __restrict__ h2c0,
                               const float* __restrict__ h2c1,
                               const float* __restrict__ dinv,
                               float* __restrict__ agg0,
                               float* __restrict__ agg1, int E) {
  int e = blockIdx.x * blockDim.x + threadIdx.x;
  if (e >= E) return;
  int a = src[e], b = dst[e];
  float nrm = dinv[a] * dinv[b];
  unsafeAtomicAdd(&agg0[b], h2c0[a] * nrm);
  unsafeAtomicAdd(&agg1[b], h2c1[a] * nrm);
}

__global__ void gcn_finalize(const float* __restrict__ agg0,
                             const float* __restrict__ agg1,
                             const float* __restrict__ h2c0,
                             const float* __restrict__ h2c1,
                             const float* __restrict__ dinv,
                             const float* __restrict__ b2,
                             float* __restrict__ out, int n) {
  int i = blockIdx.x * blockDim.x + threadIdx.x;
  if (i >= n) return;
  float di = dinv[i];
  float sl = di * di;                           // self-loop norm = 1/deg
  float z0 = agg0[i] + h2c0[i] * sl + b2[0];
  float z1 = agg1[i] + h2c1[i] * sl + b2[1];
  float mx = fmaxf(z0, z1);
  float lse = mx + logf(expf(z0 - mx) + expf(z1 - mx));
  float2 r; r.x = z0 - lse; r.y = z1 - lse;
  *(float2*)(out + (size_t)i * 2) = r;
}

extern "C" void kernel_launch(void* const* d_in, const int* in_sizes, int n_in,
                              void* d_out, int out_size, void* d_ws, size_t ws_size,
                              hipStream_t stream) {
  const float* x  = (const float*)d_in[0];      // [N,1]
  const int*   ei = (const int*)d_in[1];        // [2,E]
  const float* W1 = (const float*)d_in[2];      // [1,16]
  const float* b1 = (const float*)d_in[3];      // [16]
  const float* W2 = (const float*)d_in[4];      // [16,2]
  const float* b2 = (const float*)d_in[5];      // [2]
  float* out = (float*)d_out;                   // [N,2]

  int n = in_sizes[0];
  int E = in_sizes[1] / 2;
  const int* srcp = ei;
  const int* dstp = ei + E;

  size_t ns = ((size_t)n + 31) & ~(size_t)31;   // padded stride, 128B-aligned

  float* ws    = (float*)d_ws;
  float* deg   = ws;
  float* dinv  = ws + ns;
  float* s     = ws + 2 * ns;
  float* h2c0  = ws + 3 * ns;
  float* h2c1  = ws + 4 * ns;
  float* agg0  = ws + 5 * ns;
  float* agg1  = ws + 6 * ns;
  float* relu1 = ws + 7 * ns;

  int gN = (n + TPB - 1) / TPB;
  int gE = (E + TPB - 1) / TPB;

  gcn_init<<<gN, TPB, 0, stream>>>(deg, s, agg0, agg1, n);
  gcn_degree<<<gE, TPB, 0, stream>>>(dstp, deg, E);
  gcn_dinv<<<gN, TPB, 0, stream>>>(deg, dinv, n);
  gcn_l1_scatter<<<gE, TPB, 0, stream>>>(srcp, dstp, x, dinv, s, E);
  gcn_l1_finalize<<<gN, TPB, 0, stream>>>(x, dinv, s, W1, b1, relu1, n);

  int ntiles = (n + 15) / 16;
  int gW = (int)(((size_t)ntiles * 32 + TPB - 1) / TPB);
  gcn_l2_dense_wmma<<<gW, TPB, 0, stream>>>(relu1, W2, h2c0, h2c1, n, ntiles);

  gcn_l2_scatter<<<gE, TPB, 0, stream>>>(srcp, dstp, h2c0, h2c1, dinv,
                                         agg0, agg1, E);
  gcn_finalize<<<gN, TPB, 0, stream>>>(agg0, agg1, h2c0, h2c1, dinv, b2, out, n);
}